// UmbrellaSurfaceConstructorV2_83588653515413
// MI455X (gfx1250) — compile-verified
//
#include <hip/hip_runtime.h>
#include <hip/hip_bf16.h>
#include <math.h>

// ---------------- problem constants ----------------
#define BATCH 8
#define NPTS 2048
#define K1 21
#define K2C 3
#define K3C 5
#define KK 4                         // K3-1
#define CCH 10
#define CPAD 12                      // padded channel rows (K loop reaches k=11)
#define SCOLS (KK * K2C * NPTS)      // 24576 columns per batch
#define PPOS (BATCH * SCOLS)         // 196608 = BN reduction count
#define PI_F 3.14159265358979323846f

typedef float v2f __attribute__((ext_vector_type(2)));
typedef float v8f __attribute__((ext_vector_type(8)));
typedef unsigned int v4u __attribute__((ext_vector_type(4)));
typedef unsigned int v8u __attribute__((ext_vector_type(8)));

// =====================================================================
// Tensor Data Mover: async 2-D tile (rows x cols of f32) Global -> LDS.
// 2-group descriptor form (tensors up to 2D), per CDNA5 ISA.
// Issued by one wave; waits TENSORcnt==0 before returning.
// =====================================================================
__device__ __forceinline__ void tdm_load_2d_f32(unsigned lds_off, const void* gptr,
                                                unsigned dim0, unsigned dim1,
                                                unsigned tile0, unsigned tile1,
                                                unsigned stride0) {
    unsigned long long ga = (unsigned long long)gptr;
    v4u g0;
    g0.x = 1u;                                   // count=1, user descriptor
    g0.y = lds_off;                              // lds_addr (bytes)
    g0.z = (unsigned)(ga & 0xffffffffu);         // global_addr[31:0]
    g0.w = (unsigned)(ga >> 32) | (2u << 30);    // global_addr[56:32] | type=2
    v8u g1;
    g1.s0 = (2u << 16);                          // workgroup_mask=0, data_size=2 (4B)
    g1.s1 = (dim0 & 0xffffu) << 16;              // tensor_dim0[15:0] @ bit48
    g1.s2 = (dim0 >> 16) | ((dim1 & 0xffffu) << 16);
    g1.s3 = (dim1 >> 16) | (tile0 << 16);        // tile_dim0
    g1.s4 = (tile1 & 0xffffu);                   // tile_dim1, tile_dim2=0
    g1.s5 = stride0;                             // tensor_dim0_stride[31:0]
    g1.s6 = 0u;
    g1.s7 = 0u;
    asm volatile("tensor_load_to_lds %0, %1\n\t"
                 "s_wait_tensorcnt 0x0"
                 :: "s"(g0), "s"(g1) : "memory");
}

// =====================================================================
// Kernel 1: KNN top-21 + FPS -> K2 umbrella centers per point.
// Batch cloud [3][2048] staged to LDS by the TDM (24KB), sq-norms derived.
// =====================================================================
__global__ void knn_fps_kernel(const float* __restrict__ center, float* __restrict__ centout) {
    __shared__ float sp[3 * NPTS];       // [3][2048]: x row, y row, z row
    __shared__ float ssq[NPTS];
    const int b  = blockIdx.x >> 3;      // 8 blocks per batch
    const int nb = blockIdx.x & 7;
    const float* cb = center + (size_t)b * 3 * NPTS;
    if (threadIdx.x < 32)
        tdm_load_2d_f32((unsigned)(uintptr_t)(&sp[0]), cb, NPTS, 3, NPTS, 3, NPTS);
    __syncthreads();
    for (int i = threadIdx.x; i < NPTS; i += 256) {
        float x = sp[i], y = sp[NPTS + i], z = sp[2 * NPTS + i];
        ssq[i] = x * x + y * y + z * z;
    }
    __syncthreads();

    const int n = nb * 256 + threadIdx.x;
    const float px = sp[n], py = sp[NPTS + n], pz = sp[2 * NPTS + n], psq = ssq[n];

    float bestd[K1]; int besti[K1];
    #pragma unroll
    for (int i = 0; i < K1; ++i) { bestd[i] = 3.0e38f; besti[i] = 0; }
    for (int m = 0; m < NPTS; ++m) {
        float d = psq + ssq[m] - 2.0f * (px * sp[m] + py * sp[NPTS + m] + pz * sp[2 * NPTS + m]);
        if (d < bestd[K1 - 1]) {
            int j = K1 - 1;
            while (j > 0 && d < bestd[j - 1]) { bestd[j] = bestd[j - 1]; besti[j] = besti[j - 1]; --j; }
            bestd[j] = d; besti[j] = m;
        }
    }
    // neighbors 1..20 (drop self), then deterministic FPS -> 3 centers
    float gx[K1 - 1], gy[K1 - 1], gz[K1 - 1], dist[K1 - 1];
    for (int i = 0; i < K1 - 1; ++i) {
        int m = besti[i + 1];
        gx[i] = sp[m]; gy[i] = sp[NPTS + m]; gz[i] = sp[2 * NPTS + m]; dist[i] = 1.0e10f;
    }
    int last = 0;
    float cx[K2C], cy[K2C], cz[K2C];
    for (int t = 0; t < K2C; ++t) {
        cx[t] = gx[last]; cy[t] = gy[last]; cz[t] = gz[last];
        const float lx = gx[last], ly = gy[last], lz = gz[last];
        float best = -1.0f; int bi = 0;
        for (int i = 0; i < K1 - 1; ++i) {
            float dx = gx[i] - lx, dy = gy[i] - ly, dz = gz[i] - lz;
            float d = dx * dx + dy * dy + dz * dz;
            dist[i] = fminf(dist[i], d);
            if (dist[i] > best) { best = dist[i]; bi = i; }   // first-max == jnp.argmax
        }
        last = bi;
    }
    float* cp = centout + ((size_t)(b * NPTS + n)) * (K2C * 3);
    #pragma unroll
    for (int t = 0; t < K2C; ++t) { cp[t * 3 + 0] = cx[t]; cp[t * 3 + 1] = cy[t]; cp[t * 3 + 2] = cz[t]; }
}

// =====================================================================
// Kernel 2: per (b,n,k2): top-5 KNN of center, drop self, sort by phi,
// umbrella triangles -> feat, written into [B, CPAD, kk, k2*N] layout
// (rows 10,11 written as zeros for the padded GEMM).
// =====================================================================
__global__ void knn2_feat_kernel(const float* __restrict__ center, const float* __restrict__ centin,
                                 float* __restrict__ Xout) {
    __shared__ float sp[3 * NPTS];
    __shared__ float ssq[NPTS];
    const int b   = blockIdx.x / 24;     // 24 blocks per batch (6144 queries)
    const int qb0 = (blockIdx.x % 24) * 256;
    const float* cb = center + (size_t)b * 3 * NPTS;
    if (threadIdx.x < 32)
        tdm_load_2d_f32((unsigned)(uintptr_t)(&sp[0]), cb, NPTS, 3, NPTS, 3, NPTS);
    __syncthreads();
    for (int i = threadIdx.x; i < NPTS; i += 256) {
        float x = sp[i], y = sp[NPTS + i], z = sp[2 * NPTS + i];
        ssq[i] = x * x + y * y + z * z;
    }
    __syncthreads();

    const int q = qb0 + threadIdx.x;     // 0..6143
    const int n = q / K2C;
    const int t = q - n * K2C;
    const float* cp = centin + ((size_t)(b * NPTS + n)) * (K2C * 3) + t * 3;
    const float cxx = cp[0], cyy = cp[1], czz = cp[2];
    const float csq = cxx * cxx + cyy * cyy + czz * czz;

    float bestd[K3C]; int besti[K3C];
    #pragma unroll
    for (int i = 0; i < K3C; ++i) { bestd[i] = 3.0e38f; besti[i] = 0; }
    for (int m = 0; m < NPTS; ++m) {
        float d = csq + ssq[m] - 2.0f * (cxx * sp[m] + cyy * sp[NPTS + m] + czz * sp[2 * NPTS + m]);
        if (d < bestd[K3C - 1]) {
            int j = K3C - 1;
            while (j > 0 && d < bestd[j - 1]) { bestd[j] = bestd[j - 1]; besti[j] = besti[j - 1]; --j; }
            bestd[j] = d; besti[j] = m;
        }
    }
    float rx[KK], ry[KK], rz[KK], ph[KK];
    #pragma unroll
    for (int i = 0; i < KK; ++i) {
        int m = besti[i + 1];            // drop self (rank 0)
        rx[i] = sp[m] - cxx; ry[i] = sp[NPTS + m] - cyy; rz[i] = sp[2 * NPTS + m] - czz;
        ph[i] = atan2f(ry[i], rx[i]);
    }
    // stable argsort by phi (rank counting)
    float ax[KK], ay[KK], az[KK];
    #pragma unroll
    for (int i = 0; i < KK; ++i) {
        int r = 0;
        #pragma unroll
        for (int j = 0; j < KK; ++j)
            if (ph[j] < ph[i] || (ph[j] == ph[i] && j < i)) ++r;
        ax[r] = rx[i]; ay[r] = ry[i]; az[r] = rz[i];
    }
    float* xb = Xout + (size_t)b * CPAD * SCOLS + (size_t)t * NPTS + n;
    float sgn = 1.0f;
    const float third = 1.0f / 3.0f;
    for (int j = 0; j < KK; ++j) {
        const int j2 = (j + 1) & 3;
        const float v1x = ax[j],  v1y = ay[j],  v1z = az[j];
        const float v2x = ax[j2], v2y = ay[j2], v2z = az[j2];
        float nx = v1y * v2z - v1z * v2y;
        float ny = v1z * v2x - v1x * v2z;
        float nz = v1x * v2y - v1y * v2x;
        const float len = sqrtf(nx * nx + ny * ny + nz * nz);
        const float inv = 1.0f / fmaxf(len, 1e-12f);
        const float ux = nx * inv, uy = ny * inv, uz = nz * inv;
        if (j == 0) sgn = (ux > 0.0f) ? 1.0f : -1.0f;
        float nmx = ux * sgn, nmy = uy * sgn, nmz = uz * sgn;
        const float gx3 = (v1x + v2x) * third, gy3 = (v1y + v2y) * third, gz3 = (v1z + v2z) * third;
        const float rho = sqrtf(gx3 * gx3 + gy3 * gy3 + gz3 * gz3);
        const float th  = acosf(fminf(fmaxf(gz3 / fmaxf(rho, 1e-8f), -1.0f), 1.0f));
        const float ph2 = atan2f(gy3, gx3);
        float pos = (nmx * gx3 + nmy * gy3 + nmz * gz3) * 0.5773502691896258f;
        if (__builtin_isnan(nmx) || __builtin_isnan(nmy) || __builtin_isnan(nmz)) {
            nmx = 0.0f; nmy = 0.0f; nmz = 0.0f; pos = 0.0f;
        }
        const float f[CPAD] = { gx3, gy3, gz3,
                                rho, th * (1.0f / PI_F), ph2 * (1.0f / (2.0f * PI_F)) + 0.5f,
                                nmx, nmy, nmz, pos, 0.0f, 0.0f };
        #pragma unroll
        for (int c = 0; c < CPAD; ++c)
            xb[((size_t)c * KK + j) * (K2C * NPTS)] = f[c];
    }
}

// =====================================================================
// GEMM: Y[b] = W(10x10, LDS-padded to 16x12) @ X[b](12 x S, rows 10-11 zero)
// HAS_BN variant fuses relu(a[k]*x+sh[k]) on the input with LDS-staged,
// zero-padded BN coefficients — compile-time flag, so the K loop is fully
// branch-free straight-line code feeding 3x WMMA f32 16x16x4 per tile.
// =====================================================================
template <bool HAS_BN>
__global__ void gemm10_wmma_kernel(const float* __restrict__ X, const float* __restrict__ W,
                                   const float* __restrict__ bias, const float* __restrict__ bnc,
                                   float* __restrict__ Y) {
    __shared__ float sW[16 * CPAD];      // A zero-padded: [m=16][k=12]
    __shared__ float sBN[2 * CPAD];      // a[12], sh[12]
    __shared__ float sB[CPAD];           // bias padded with zeros
    const int tid = threadIdx.x;
    for (int i = tid; i < 16 * CPAD; i += 256) {
        const int m = i / CPAD, k = i - m * CPAD;
        sW[i] = (m < CCH && k < CCH) ? W[m * CCH + k] : 0.0f;
    }
    if (HAS_BN && tid < 2 * CPAD) sBN[tid] = bnc[tid];
    if (tid < CPAD) sB[tid] = (bias && tid < CCH) ? bias[tid] : 0.0f;
    __syncthreads();

    const int lane = tid & 31;
    const int wave = tid >> 5;
    const int tile = blockIdx.x * (blockDim.x >> 5) + wave;   // 0..12287, exact grid
    const int g0   = tile * 16;
    const int b    = g0 / SCOLS;
    const int col0 = g0 - b * SCOLS;
    const int mr   = lane & 15;          // M row of A / N col of B,C
    const int kh   = lane >> 4;          // half-wave: K pair for A/B, M+8 for C/D

    const float* Xb = X + (size_t)b * CPAD * SCOLS + col0 + mr;
    v8f c = {0.f, 0.f, 0.f, 0.f, 0.f, 0.f, 0.f, 0.f};
    #pragma unroll
    for (int kc = 0; kc < 3; ++kc) {
        v2f av, bv;
        #pragma unroll
        for (int v = 0; v < 2; ++v) {
            const int k = kc * 4 + kh * 2 + v;       // 0..11, always in padded range
            av[v] = sW[mr * CPAD + k];
            float xv = Xb[(size_t)k * SCOLS];
            if (HAS_BN) xv = fmaxf(sBN[k] * xv + sBN[CPAD + k], 0.0f);
            bv[v] = xv;
        }
        c = __builtin_amdgcn_wmma_f32_16x16x4_f32(false, av, false, bv, (short)0, c, false, false);
    }
    float* Yb = Y + (size_t)b * CPAD * SCOLS + col0 + mr;
    #pragma unroll
    for (int v = 0; v < 4; ++v) {        // rows m = v (kh=0) / v+8 (kh=1): always < 12
        const int m = v + 8 * kh;
        Yb[(size_t)m * SCOLS] = c[v] + sB[m];
    }
    if (kh == 0) {
        #pragma unroll
        for (int v = 4; v < 8; ++v)      // rows 4..7 (kh=1 side would be 12..15: dropped)
            Yb[(size_t)v * SCOLS] = c[v] + sB[v];
    }
}

// =====================================================================
// Deterministic BN statistics: fixed 48-block grid, register accumulators,
// fixed-order LDS tree reduce; then one block folds partials into padded
// scale/shift coefficients a[12], sh[12].
// =====================================================================
__global__ void stats_partial_kernel(const float* __restrict__ Y, float* __restrict__ part) {
    __shared__ float s_acc[256][2 * CCH];    // 20 KB
    float acc[2 * CCH];
    #pragma unroll
    for (int k = 0; k < 2 * CCH; ++k) acc[k] = 0.0f;
    for (int p = blockIdx.x * blockDim.x + threadIdx.x; p < PPOS; p += gridDim.x * blockDim.x) {
        const int b = p / SCOLS;
        const int col = p - b * SCOLS;
        const float* yb = Y + (size_t)b * CPAD * SCOLS + col;
        #pragma unroll
        for (int k = 0; k < CCH; ++k) {
            float v = yb[(size_t)k * SCOLS];
            acc[k] += v; acc[CCH + k] += v * v;
        }
    }
    #pragma unroll
    for (int k = 0; k < 2 * CCH; ++k) s_acc[threadIdx.x][k] = acc[k];
    __syncthreads();
    for (int s = 128; s > 0; s >>= 1) {
        if (threadIdx.x < s) {
            #pragma unroll
            for (int k = 0; k < 2 * CCH; ++k) s_acc[threadIdx.x][k] += s_acc[threadIdx.x + s][k];
        }
        __syncthreads();
    }
    if (threadIdx.x < 2 * CCH) part[blockIdx.x * (2 * CCH) + threadIdx.x] = s_acc[0][threadIdx.x];
}

__global__ void stats_to_bncoef_kernel(const float* __restrict__ part,
                                       const float* __restrict__ gamma,
                                       const float* __restrict__ beta,
                                       float* __restrict__ bnc) {
    const int t = threadIdx.x;
    if (t < CPAD) {
        float a = 0.0f, sh = 0.0f;
        if (t < CCH) {
            float sum = 0.0f, sq = 0.0f;
            for (int i = 0; i < 48; ++i) {
                sum += part[i * (2 * CCH) + t];
                sq  += part[i * (2 * CCH) + CCH + t];
            }
            const float inv_cnt = 1.0f / (float)PPOS;
            float mu  = sum * inv_cnt;
            float var = sq * inv_cnt - mu * mu;       // biased, matches reference
            a  = gamma[t] * rsqrtf(var + 1e-5f);
            sh = beta[t] - mu * a;
        }
        bnc[t] = a; bnc[CPAD + t] = sh;
    }
}

// =====================================================================
// Final aggregation: max over kk=4, mean over k2=3 -> out [B,10,N]
// =====================================================================
__global__ void final_aggr_kernel(const float* __restrict__ Z, float* __restrict__ out) {
    const int idx = blockIdx.x * blockDim.x + threadIdx.x;   // B*10*N = 163840
    if (idx >= BATCH * CCH * NPTS) return;
    const int n  = idx & (NPTS - 1);
    const int bc = idx >> 11;                                 // b*10 + c
    const int b  = bc / CCH;
    const int cc = bc - b * CCH;
    const float* zb = Z + ((size_t)b * CPAD + cc) * SCOLS;
    float s = 0.0f;
    #pragma unroll
    for (int t = 0; t < K2C; ++t) {
        float m = -3.0e38f;
        #pragma unroll
        for (int j = 0; j < KK; ++j)
            m = fmaxf(m, zb[(size_t)j * (K2C * NPTS) + t * NPTS + n]);
        s += m;
    }
    out[idx] = s * (1.0f / 3.0f);
}

// =====================================================================
extern "C" void kernel_launch(void* const* d_in, const int* in_sizes, int n_in,
                              void* d_out, int out_size, void* d_ws, size_t ws_size,
                              hipStream_t stream) {
    const float* center = (const float*)d_in[0];
    const float* w1  = (const float*)d_in[1];
    const float* g1  = (const float*)d_in[2];
    const float* be1 = (const float*)d_in[3];
    const float* w2  = (const float*)d_in[4];
    const float* b2  = (const float*)d_in[5];
    const float* g2  = (const float*)d_in[6];
    const float* be2 = (const float*)d_in[7];
    const float* w3  = (const float*)d_in[8];
    const float* b3  = (const float*)d_in[9];
    float* out = (float*)d_out;

    // workspace layout (floats)
    const size_t XSZ = (size_t)BATCH * CPAD * SCOLS;   // 2,359,296 per activation buffer
    float* ws      = (float*)d_ws;
    float* xbuf    = ws;                  // feat / y2
    float* ybuf    = xbuf + XSZ;          // y1 / z
    float* centbuf = ybuf + XSZ;          // [B,N,K2,3] = 147,456
    float* bnc1    = centbuf + (size_t)BATCH * NPTS * K2C * 3;
    float* bnc2    = bnc1 + 2 * CPAD;
    float* part    = bnc2 + 2 * CPAD;     // 48 * 20 partials

    // geometry stage (TDM-staged LDS point cloud)
    knn_fps_kernel  <<<BATCH * 8, 256, 0, stream>>>(center, centbuf);
    knn2_feat_kernel<<<BATCH * 24, 256, 0, stream>>>(center, centbuf, xbuf);

    // MLP stage: 12288 WMMA tiles per GEMM, 8 waves/block -> 1536 blocks exact
    gemm10_wmma_kernel<false><<<1536, 256, 0, stream>>>(xbuf, w1, nullptr, nullptr, ybuf);
    stats_partial_kernel<<<48, 256, 0, stream>>>(ybuf, part);
    stats_to_bncoef_kernel<<<1, 32, 0, stream>>>(part, g1, be1, bnc1);
    gemm10_wmma_kernel<true><<<1536, 256, 0, stream>>>(ybuf, w2, b2, bnc1, xbuf);
    stats_partial_kernel<<<48, 256, 0, stream>>>(xbuf, part);
    stats_to_bncoef_kernel<<<1, 32, 0, stream>>>(part, g2, be2, bnc2);
    gemm10_wmma_kernel<true><<<1536, 256, 0, stream>>>(xbuf, w3, b3, bnc2, ybuf);

    final_aggr_kernel<<<(BATCH * CCH * NPTS + 255) / 256, 256, 0, stream>>>(ybuf, out);
}